// Model_87333864997456
// MI455X (gfx1250) — compile-verified
//
#include <hip/hip_runtime.h>
#include <hip/hip_bf16.h>

typedef __attribute__((ext_vector_type(2))) float v2f;
typedef __attribute__((ext_vector_type(4))) float v4f;
typedef __attribute__((ext_vector_type(8))) float v8f;

#define R_DIM      64      // max LoRA rank (K dimension)
#define TILE_M     16      // tokens per M-tile (one WMMA tile)
#define WAVE_N     64      // N columns per wave = 4 WMMA tiles
#define BLOCK_N    512     // 8 waves * 64 columns
#define LDS_STRIDE 68      // 64 + 4 pad: conflict-free ds_load_b64, keeps 16B align

__global__ __launch_bounds__(256) void lora_wmma_f32_kernel(
    const float* __restrict__ x,
    const float* __restrict__ weights,        // [L, N, R]
    const int*   __restrict__ seg_indptr,     // [bs+1]
    const int*   __restrict__ weight_indices, // [bs]
    const int*   __restrict__ lora_ranks,     // [L]
    const float* __restrict__ scalings,       // [L]
    const float* __restrict__ base,           // [S, N]
    float*       __restrict__ out,            // [S, N]
    int N, int bs)
{
    __shared__ float lds_x[TILE_M * LDS_STRIDE];

    const int m0 = blockIdx.y * TILE_M;

    // ---- uniform segment -> adapter lookup (depends only on blockIdx) ----
    int seg = 0;
    while (seg + 1 < bs && seg_indptr[seg + 1] <= m0) ++seg;
    const int   w_idx = weight_indices[seg];
    const int   rank  = lora_ranks[w_idx];
    const float scal  = scalings[w_idx];

    // ---- stage x tile into LDS: rank-masked and pre-scaled ----
    {
        const int t  = threadIdx.x;        // 0..255
        const int r  = t >> 4;             // row 0..15
        const int c4 = (t & 15) << 2;      // col 0,4,...,60
        const float* xp = x + (size_t)(m0 + r) * R_DIM + c4;
        v4f v;
        v.x = (c4 + 0 < rank) ? xp[0] * scal : 0.0f;
        v.y = (c4 + 1 < rank) ? xp[1] * scal : 0.0f;
        v.z = (c4 + 2 < rank) ? xp[2] * scal : 0.0f;
        v.w = (c4 + 3 < rank) ? xp[3] * scal : 0.0f;
        *(v4f*)(&lds_x[r * LDS_STRIDE + c4]) = v;
    }
    __syncthreads();

    const int wave = threadIdx.x >> 5;
    const int lane = threadIdx.x & 31;
    const int half = lane >> 4;   // ISA: lanes 16-31 carry K+2,K+3 for A/B frags
    const int l16  = lane & 15;

    const int n0 = blockIdx.x * BLOCK_N + wave * WAVE_N;

    // ---- rank-0 adapter: out = base, pure NT streaming copy (block-uniform) ----
    if (rank == 0) {
        #pragma unroll
        for (int nt = 0; nt < 4; ++nt) {
            const int ncol = n0 + nt * 16;
            const float* cp = base + (size_t)(m0 + half * 8) * N + ncol + l16;
            float*       op = out  + (size_t)(m0 + half * 8) * N + ncol + l16;
            #pragma unroll
            for (int j = 0; j < 8; ++j) {
                float v = __builtin_nontemporal_load(cp + (size_t)j * N);
                __builtin_nontemporal_store(v, op + (size_t)j * N);
            }
        }
        return;
    }

    // ---- preload all 16 A fragments (K steps of 4) from LDS ----
    v2f afrag[16];
    {
        const float* ap = &lds_x[l16 * LDS_STRIDE + half * 2];
        #pragma unroll
        for (int kk = 0; kk < 16; ++kk)
            afrag[kk] = *(const v2f*)(ap + kk * 4);   // ds_load_b64
    }

    #pragma unroll
    for (int nt = 0; nt < 4; ++nt) {
        const int ncol = n0 + nt * 16;

        // ---- load C tile from base_output, non-temporal (single-use stream)
        //      C layout: VGPR j -> M = j + 8*half, N = l16
        v8f acc;
        {
            const float* cp = base + (size_t)(m0 + half * 8) * N + ncol + l16;
            #pragma unroll
            for (int j = 0; j < 8; ++j)
                acc[j] = __builtin_nontemporal_load(cp + (size_t)j * N);
        }

        // ---- B fragments straight from global (L2-resident weights, RT hint) ----
        // B column n == weights row (ncol + l16); k pair = kk*4 + 2*half
        const float* wp = weights + ((size_t)w_idx * N + (ncol + l16)) * R_DIM + half * 2;

        #pragma unroll
        for (int kk = 0; kk < 16; ++kk) {
            v2f b = *(const v2f*)(wp + kk * 4);       // global_load_b64
            acc = __builtin_amdgcn_wmma_f32_16x16x4_f32(
                false, afrag[kk], false, b, (short)0, acc, false, false);
        }

        // ---- store D tile, non-temporal ----
        {
            float* op = out + (size_t)(m0 + half * 8) * N + ncol + l16;
            #pragma unroll
            for (int j = 0; j < 8; ++j)
                __builtin_nontemporal_store(acc[j], op + (size_t)j * N);
        }
    }
}

extern "C" void kernel_launch(void* const* d_in, const int* in_sizes, int n_in,
                              void* d_out, int out_size, void* d_ws, size_t ws_size,
                              hipStream_t stream) {
    const float* x              = (const float*)d_in[0];
    const float* weights        = (const float*)d_in[1];
    const int*   seg_indptr     = (const int*)d_in[5];
    const int*   weight_indices = (const int*)d_in[6];
    const int*   lora_ranks     = (const int*)d_in[7];
    const float* scalings       = (const float*)d_in[8];
    const float* base           = (const float*)d_in[12];
    float*       out            = (float*)d_out;

    const int bs = in_sizes[6];          // number of segments
    const int L  = in_sizes[7];          // number of adapters
    const int N  = in_sizes[1] / (L * R_DIM);
    const int S  = in_sizes[0] / R_DIM;

    dim3 grid(N / BLOCK_N, S / TILE_M);  // (8, 1024) for the given shapes
    dim3 block(256);
    lora_wmma_f32_kernel<<<grid, block, 0, stream>>>(
        x, weights, seg_indptr, weight_indices, lora_ranks, scalings,
        base, out, N, bs);
}